// ConvertParamsLayer_16277926052102
// MI455X (gfx1250) — compile-verified
//
#include <hip/hip_runtime.h>

typedef __attribute__((ext_vector_type(2))) float v2f;
typedef __attribute__((ext_vector_type(8))) float v8f;

#define HDIM 64
#define VDIM 256

// One block per sample b. 256 threads = 8 wave32.
// Each wave runs two chains of 16 x V_WMMA_F32_16X16X4_F32 (K=4 each -> K=64 total),
// covering a 16-wide v-slice per chain; 8 waves * 2 chains * 16 v = 256 = VDIM.
// A-tile = wt1^T [16 v x 4 h] (lane = v, k-slot = h); B = coeff[h] broadcast over N.
// The same loaded wt1 element is scaled and streamed out to wt2 (non-temporal),
// so wt1 is read exactly once and wt2 written exactly once (bandwidth-optimal).
__global__ __launch_bounds__(256) void convert_params_wmma(
    const float* __restrict__ b1,
    const float* __restrict__ wt1,
    const float* __restrict__ muh1,
    const float* __restrict__ muh2,
    const float* __restrict__ cov1,
    const float* __restrict__ cov2,
    float* __restrict__ b2,
    float* __restrict__ wt2)
{
    __shared__ float s_scale[HDIM];
    __shared__ float s_coeff[HDIM];

    const int b   = blockIdx.x;
    const int tid = threadIdx.x;

    if (tid < HDIM) {
        const float c1 = cov1[(size_t)b * HDIM + tid];
        const float c2 = cov2[(size_t)b * HDIM + tid];
        const float sc = sqrtf(c1 / c2);
        s_scale[tid] = sc;
        s_coeff[tid] = muh1[(size_t)b * HDIM + tid] - sc * muh2[(size_t)b * HDIM + tid];
    }
    __syncthreads();

    const int lane = tid & 31;
    const int wave = tid >> 5;         // 0..7
    const int lo   = lane & 15;        // v offset within 16-wide tile
    const int hsel = (lane >> 4) * 2;  // lanes 0-15 -> k=0,1 ; lanes 16-31 -> k=2,3

    const float* __restrict__ wt1b = wt1 + (size_t)b * HDIM * VDIM;
    float* __restrict__       wt2b = wt2 + (size_t)b * HDIM * VDIM;

    for (int chain = 0; chain < 2; ++chain) {
        const int v0 = ((wave << 1) + chain) << 4;   // 16-wide v slice
        v8f acc = {};                                 // C starts at zero

#pragma unroll 4
        for (int h0 = 0; h0 < HDIM; h0 += 4) {
            const int ha = h0 + hsel;       // row for A slot (vgpr0, this lane half)
            const int hb = ha + 1;          // row for A slot (vgpr1, this lane half)

            const float* pa = wt1b + (size_t)ha * VDIM + v0 + lo;
            const float* pb = wt1b + (size_t)hb * VDIM + v0 + lo;

            const float a0 = __builtin_nontemporal_load(pa);   // read-once stream
            const float a1 = __builtin_nontemporal_load(pb);

            // speculative prefetch 2 k-steps (8 rows) ahead -> global_prefetch_b8
            __builtin_prefetch(pa + 8 * VDIM, 0, 0);

            v2f av = { a0, a1 };
            v2f bv = { s_coeff[ha], s_coeff[hb] };   // broadcast over N columns

            // D = A(16x4, f32) x B(4x16, f32) + C ; 8 args: neg_a, A, neg_b, B, c_mod, C, reuse_a, reuse_b
            acc = __builtin_amdgcn_wmma_f32_16x16x4_f32(
                false, av, false, bv, (short)0, acc, false, false);

            // fold the wt2 = scale * wt1 stream into the same pass (non-temporal)
            __builtin_nontemporal_store(s_scale[ha] * a0,
                                        wt2b + (size_t)ha * VDIM + v0 + lo);
            __builtin_nontemporal_store(s_scale[hb] * a1,
                                        wt2b + (size_t)hb * VDIM + v0 + lo);
        }

        // D[m,n] is identical for every n (B was column-broadcast).
        // C/D layout: lane 0 holds M=0..7 in VGPRs 0..7, lane 16 holds M=8..15.
        if (lo == 0) {
            const int vbase = v0 + (lane >> 4) * 8;
            const float* b1p = b1 + (size_t)b * VDIM + vbase;
            float*       b2p = b2 + (size_t)b * VDIM + vbase;
#pragma unroll
            for (int r = 0; r < 8; ++r)
                b2p[r] = b1p[r] + acc[r];
        }
    }
}

extern "C" void kernel_launch(void* const* d_in, const int* in_sizes, int n_in,
                              void* d_out, int out_size, void* d_ws, size_t ws_size,
                              hipStream_t stream) {
    (void)n_in; (void)out_size; (void)d_ws; (void)ws_size;

    const float* b1   = (const float*)d_in[0];   // [B, V]
    const float* wt1  = (const float*)d_in[1];   // [B, H, V]
    const float* muh1 = (const float*)d_in[2];   // [B, H]
    const float* muh2 = (const float*)d_in[3];   // [B, H]
    const float* cov1 = (const float*)d_in[4];   // [B, H]
    const float* cov2 = (const float*)d_in[5];   // [B, H]

    const int Bn = in_sizes[0] / VDIM;           // 4096

    float* b2  = (float*)d_out;                  // [B, V] first
    float* wt2 = b2 + (size_t)Bn * VDIM;         // [B, H, V] second

    convert_params_wmma<<<dim3(Bn), dim3(256), 0, stream>>>(
        b1, wt1, muh1, muh2, cov1, cov2, b2, wt2);
}